// SimpleVectorQuantizer_3341484556867
// MI455X (gfx1250) — compile-verified
//
#include <hip/hip_runtime.h>
#include <hip/hip_bf16.h>

// ---------------------------------------------------------------------------
// SimpleVectorQuantizer for MI455X (gfx1250, wave32, WMMA).
//
// Shapes: x [262144 x 64] f32, embeddings [64 x 512] f32 (D x K).
// Outputs (concatenated): quantized [N*64] f32, loss [1] f32, indices [N] f32.
//
// Matmul via V_WMMA_F32_16X16X4_F32 (full f32 so the argmin matches the f32
// reference). Codebook + ||e||^2 staged in LDS per block. Two independent
// WMMA accumulators per wave (two 16-code column tiles in flight) to break
// the RAW accumulation chain and reuse each A fragment twice per load.
// ---------------------------------------------------------------------------

#define NROWS   (64 * 64 * 64)      // 262144
#define DIM     64
#define KCODES  512
#define THREADS 256
#define WAVES   (THREADS / 32)      // 8 wave32 per block
#define BLOCKS  1024
#define TILES_TOTAL (NROWS / 16)    // 16384 row tiles of 16 rows
#define ITERS   (TILES_TOTAL / (BLOCKS * WAVES))   // = 2, exact

typedef __attribute__((ext_vector_type(2))) float v2f;
typedef __attribute__((ext_vector_type(4))) float v4f;
typedef __attribute__((ext_vector_type(8))) float v8f;

// LDS layout: codebook stored pair-interleaved so one B fragment is one b64:
//   eLds[(d>>1)*(2*KCODES) + 2*c + (d&1)] == E[d][c]
// followed by enorm[KCODES] and a WAVES-float reduction scratchpad.
#define ELDS_FLOATS (DIM * KCODES)          // 32768 floats = 128 KB
#define SMEM_FLOATS (ELDS_FLOATS + KCODES + WAVES)

__global__ __launch_bounds__(THREADS, 1)
void vq_main_kernel(const float* __restrict__ x,
                    const float* __restrict__ emb,     // [DIM][KCODES] row-major
                    float* __restrict__ out_q,         // [NROWS*DIM]
                    float* __restrict__ out_idx,       // [NROWS] (as float)
                    float* __restrict__ block_loss)    // [BLOCKS]
{
    extern __shared__ float smem[];
    float* eLds  = smem;                       // pair-interleaved codebook
    float* enorm = smem + ELDS_FLOATS;         // ||e_k||^2
    float* redbf = enorm + KCODES;             // per-wave loss partials

    const int tid = threadIdx.x;

    // ---- Stage codebook into LDS (pair-interleaved) ----------------------
    for (int i = tid; i < DIM * KCODES; i += THREADS) {
        const int d = i >> 9;                  // KCODES == 512
        const int c = i & (KCODES - 1);
        eLds[(d >> 1) * (2 * KCODES) + (c << 1) + (d & 1)] = emb[i];
    }
    __syncthreads();

    // ---- ||e_k||^2 -------------------------------------------------------
    for (int c = tid; c < KCODES; c += THREADS) {
        float s = 0.0f;
        #pragma unroll
        for (int d = 0; d < DIM; ++d) {
            const float v = eLds[(d >> 1) * (2 * KCODES) + (c << 1) + (d & 1)];
            s += v * v;
        }
        enorm[c] = s;
    }
    __syncthreads();

    const int lane = tid & 31;
    const int wave = tid >> 5;
    const int half = lane >> 4;    // 0: lanes 0-15, 1: lanes 16-31
    const int ln   = lane & 15;

    float loss_acc = 0.0f;

    for (int it = 0; it < ITERS; ++it) {
        const int tile = (it * BLOCKS + blockIdx.x) * WAVES + wave;
        const int m0   = tile * 16;            // first row of this 16-row tile

        // ---- A fragments: 16x64 tile of x held in 16 v2f per lane --------
        // ISA f32 A layout (16x4): lanes 0-15 hold K={0,1}, lanes 16-31 K={2,3}.
        v2f afr[16];
        const float* arow = x + (size_t)(m0 + ln) * DIM + half * 2;
        #pragma unroll
        for (int s = 0; s < 16; ++s)
            afr[s] = *(const v2f*)(arow + s * 4);

        float best[8];
        int   bidx[8];
        #pragma unroll
        for (int j = 0; j < 8; ++j) { best[j] = 3.0e38f; bidx[j] = 0; }

        // ---- Sweep 32 column tiles, two at a time (indep. accumulators) --
        for (int kt = 0; kt < KCODES / 16; kt += 2) {
            v8f acc0 = {};
            v8f acc1 = {};
            const int code0 = kt * 16 + ln;
            const int code1 = code0 + 16;
            const float* bb0 = eLds + (code0 << 1);
            const float* bb1 = eLds + (code1 << 1);
            #pragma unroll
            for (int s = 0; s < 16; ++s) {
                const int kpair = s * 2 + half;        // k = 4s + 2*half
                const v2f b0 = *(const v2f*)(bb0 + kpair * (2 * KCODES));
                const v2f b1 = *(const v2f*)(bb1 + kpair * (2 * KCODES));
                acc0 = __builtin_amdgcn_wmma_f32_16x16x4_f32(
                    false, afr[s], false, b0, (short)0, acc0, false, false);
                acc1 = __builtin_amdgcn_wmma_f32_16x16x4_f32(
                    false, afr[s], false, b1, (short)0, acc1, false, false);
            }
            // dist(row, code) = ||x||^2 + ||e||^2 - 2*sim ; ||x||^2 constant
            // per row, so rank by (||e||^2 - 2*sim). Candidate codes arrive
            // in ascending order per lane, so strict '<' keeps the first
            // (lowest) index among equals -- no tie-break compare needed here.
            const float en0 = enorm[code0];
            const float en1 = enorm[code1];
            #pragma unroll
            for (int j = 0; j < 8; ++j) {
                const float d0 = __builtin_fmaf(-2.0f, acc0[j], en0);
                if (d0 < best[j]) { best[j] = d0; bidx[j] = code0; }
                const float d1 = __builtin_fmaf(-2.0f, acc1[j], en1);
                if (d1 < best[j]) { best[j] = d1; bidx[j] = code1; }
            }
        }

        // ---- argmin across the 16 lanes of each half (xor 8,4,2,1) -------
        // Tie-break on lowest code index to match jnp.argmin semantics.
        #pragma unroll
        for (int j = 0; j < 8; ++j) {
            float v = best[j];
            int   i = bidx[j];
            #pragma unroll
            for (int m = 8; m >= 1; m >>= 1) {
                const float ov = __shfl_xor(v, m, 32);
                const int   oi = __shfl_xor(i, m, 32);
                if (ov < v || (ov == v && oi < i)) { v = ov; i = oi; }
            }
            bidx[j] = i;   // uniform across each 16-lane half
        }

        // ---- Emit indices, gather quantized rows, accumulate MSE ---------
        // C layout: VGPR j <-> row m0 + j + 8*half; 16 lanes of the half
        // cooperatively handle 4 floats each of the 64-float row.
        #pragma unroll
        for (int j = 0; j < 8; ++j) {
            const int row  = m0 + j + 8 * half;
            const int code = bidx[j];
            if (ln == 0) out_idx[row] = (float)code;

            const int d0 = ln * 4;                     // even, so pairs align
            const float* ep = eLds + (code << 1);
            const v2f q01 = *(const v2f*)(ep + (d0 >> 1)       * (2 * KCODES));
            const v2f q23 = *(const v2f*)(ep + ((d0 >> 1) + 1) * (2 * KCODES));
            const v4f xv  = *(const v4f*)(x + (size_t)row * DIM + d0);
            v4f qv; qv.x = q01.x; qv.y = q01.y; qv.z = q23.x; qv.w = q23.y;
            *(v4f*)(out_q + (size_t)row * DIM + d0) = qv;

            const float e0 = qv.x - xv.x, e1 = qv.y - xv.y;
            const float e2 = qv.z - xv.z, e3 = qv.w - xv.w;
            loss_acc += e0 * e0 + e1 * e1 + e2 * e2 + e3 * e3;
        }
    }

    // ---- Deterministic block reduction of the loss partial ----------------
    #pragma unroll
    for (int m = 16; m >= 1; m >>= 1)
        loss_acc += __shfl_xor(loss_acc, m, 32);
    if (lane == 0) redbf[wave] = loss_acc;
    __syncthreads();
    if (tid == 0) {
        float s = 0.0f;
        #pragma unroll
        for (int w = 0; w < WAVES; ++w) s += redbf[w];
        block_loss[blockIdx.x] = s;
    }
}

__global__ void vq_finalize_kernel(const float* __restrict__ block_loss,
                                   float* __restrict__ out_loss)
{
    if (blockIdx.x == 0 && threadIdx.x == 0) {
        float s = 0.0f;
        for (int i = 0; i < BLOCKS; ++i) s += block_loss[i];
        // loss = BETA*commitment + codebook = (0.25 + 1.0) * MSE
        out_loss[0] = 1.25f * s / (float)(NROWS * DIM);
    }
}

extern "C" void kernel_launch(void* const* d_in, const int* in_sizes, int n_in,
                              void* d_out, int out_size, void* d_ws, size_t ws_size,
                              hipStream_t stream)
{
    const float* x   = (const float*)d_in[0];   // [N, 64]
    const float* emb = (const float*)d_in[1];   // [64, 512]

    float* out      = (float*)d_out;
    float* out_q    = out;                              // N*DIM floats
    float* out_loss = out + (size_t)NROWS * DIM;        // 1 float
    float* out_idx  = out_loss + 1;                     // N floats

    float* block_loss = (float*)d_ws;                   // BLOCKS floats

    const size_t shmem = SMEM_FLOATS * sizeof(float);   // ~134 KB dynamic LDS
    hipLaunchKernelGGL(vq_main_kernel, dim3(BLOCKS), dim3(THREADS), shmem, stream,
                       x, emb, out_q, out_idx, block_loss);
    hipLaunchKernelGGL(vq_finalize_kernel, dim3(1), dim3(1), 0, stream,
                       block_loss, out_loss);
}